// IPSR_model_60790967107773
// MI455X (gfx1250) — compile-verified
//
#include <hip/hip_runtime.h>
#include <hip/hip_bf16.h>

// Problem constants (from reference)
#define HW   4096      // 64*64
#define CH   256       // channels
#define KN   3072      // known pixels
#define PM   1024      // masked pixels
#define BZ   4
#define CHW  (CH * HW) // per-image elements
#define EPSV 1e-08f
#define APAD 260       // padded A-panel row stride (floats): conflict-free ds_load_b64

typedef float v2f __attribute__((ext_vector_type(2)));
typedef float v8f __attribute__((ext_vector_type(8)));

// ---------------------------------------------------------------------------
// K0: per-column inverse norms  invn[b][p] = 1/(||inp[b,:,p]|| + eps)
// ---------------------------------------------------------------------------
__global__ void k0_norms(const float* __restrict__ inp, float* __restrict__ invn) {
    int idx = blockIdx.x * blockDim.x + threadIdx.x;     // 4*4096 threads
    if (idx >= BZ * HW) return;
    int b = idx / HW, p = idx % HW;
    const float* base = inp + (size_t)b * CHW + p;
    float s = 0.f;
    #pragma unroll 8
    for (int c = 0; c < CH; ++c) {
        float v = base[(size_t)c * HW];
        s += v * v;
    }
    invn[idx] = 1.f / (sqrtf(s) + EPSV);
}

// ---------------------------------------------------------------------------
// K1: pack normalized known, c-fastest:  knT[b][k][c] = inp[b,c,known_idx[k]]*invn
// ---------------------------------------------------------------------------
__global__ void k1_pack_knT(const float* __restrict__ inp, const float* __restrict__ invn,
                            const int* __restrict__ known_idx, float* __restrict__ knT) {
    int idx = blockIdx.x * blockDim.x + threadIdx.x;     // 4*3072*256
    if (idx >= BZ * KN * CH) return;
    int c = idx % CH;
    int k = (idx / CH) % KN;
    int b = idx / (CH * KN);
    int kp = known_idx[k];
    knT[idx] = inp[(size_t)b * CHW + (size_t)c * HW + kp] * invn[b * HW + kp];
}

// ---------------------------------------------------------------------------
// K1b: transpose reference, c-fastest:  rfT[b][p][c] = rf[b][c][p]
// ---------------------------------------------------------------------------
__global__ void k1b_pack_rfT(const float* __restrict__ rf, float* __restrict__ rfT) {
    int idx = blockIdx.x * blockDim.x + threadIdx.x;     // 4*4096*256 (p fastest -> coalesced reads)
    if (idx >= BZ * HW * CH) return;
    int p = idx % HW;
    int c = (idx / HW) % CH;
    int b = idx / (HW * CH);
    rfT[(size_t)b * CHW + (size_t)p * CH + c] = rf[(size_t)b * CHW + (size_t)c * HW + p];
}

// ---------------------------------------------------------------------------
// K2: GEMM1 + fused argmax.  sim[p,k] = sum_c rf[c,p]*kn[c,k]
// One wave owns a 16-row p-block.  The A panel (16x256) is staged once into
// a padded LDS slice (conflict-free ds_load_b64), then the wave sweeps all
// 192 k-tiles: per step 1 ds_load_b64 (A) + 1 global_load_b64 (B) + 1 WMMA.
// 64-thread blocks (2 waves) keep static LDS at 33.3 KB.
// ---------------------------------------------------------------------------
__global__ void __launch_bounds__(64) k2_sim_argmax(
        const float* __restrict__ rfT, const float* __restrict__ knT,
        float* __restrict__ vmax, int* __restrict__ ind) {
    __shared__ float apanel[2 * 16 * APAD];              // 33280 B

    int wib  = threadIdx.x >> 5;                         // wave in block (0..1)
    int lane = threadIdx.x & 31;
    int wave = blockIdx.x * 2 + wib;                     // 1024 waves
    int b    = wave >> 8;                                // 256 p-blocks per image
    int p0   = (wave & 255) * 16;

    const float* A = rfT + (size_t)b * CHW;              // rfT[p*CH + c]
    const float* B = knT + (size_t)b * KN * CH;          // knT[k*CH + c]

    // Stage this wave's A panel into LDS (16 rows x 256 floats, padded rows)
    float* ap = apanel + wib * 16 * APAD;
    for (int e = lane; e < 16 * 64; e += 32) {           // 32 float4 copies/lane
        int r = e >> 6, g = e & 63;
        float4 v = *(const float4*)&A[(size_t)(p0 + r) * CH + 4 * g];
        *(float4*)&ap[r * APAD + 4 * g] = v;
    }
    __syncthreads();

    int lr = lane & 15;
    int ca = (lane < 16) ? 0 : 2;                        // K-slot pair of this half
    const float* arow = ap + lr * APAD;                  // lane's padded A row (LDS)

    float best[8];
    int   besti[8];
    #pragma unroll
    for (int i = 0; i < 8; ++i) { best[i] = -3.402823e38f; besti[i] = 0; }

    for (int k0 = 0; k0 < KN; k0 += 16) {
        const float* brow = B + (size_t)(k0 + lr) * CH;
        v8f acc = {};
        #pragma unroll
        for (int ci = 0; ci < 64; ++ci) {
            v2f a  = *(const v2f*)&arow[4 * ci + ca];    // ds_load_b64
            v2f bb = *(const v2f*)&brow[4 * ci + ca];    // global_load_b64
            acc = __builtin_amdgcn_wmma_f32_16x16x4_f32(
                false, a, false, bb, (short)0, acc, false, false);
        }
        int kcol = k0 + lr;
        #pragma unroll
        for (int i = 0; i < 8; ++i) {                    // strict > keeps first max (low k)
            float v = acc[i];
            if (v > best[i]) { best[i] = v; besti[i] = kcol; }
        }
    }

    // cross-lane argmax reduce within each 16-lane half (rows stay in-half)
    #pragma unroll
    for (int i = 0; i < 8; ++i) {
        float v = best[i]; int ki = besti[i];
        #pragma unroll
        for (int m = 8; m >= 1; m >>= 1) {
            float ov = __shfl_xor(v, m, 32);
            int   oi = __shfl_xor(ki, m, 32);
            if (ov > v || (ov == v && oi < ki)) { v = ov; ki = oi; }
        }
        best[i] = v; besti[i] = ki;
    }
    if (lr == 0) {
        int rbase = p0 + ((lane < 16) ? 0 : 8);
        #pragma unroll
        for (int i = 0; i < 8; ++i) {
            vmax[b * HW + rbase + i] = best[i];
            ind [b * HW + rbase + i] = besti[i];
        }
    }
}

// ---------------------------------------------------------------------------
// K3: sequential scan over masked pixels (one block of 256 threads per image;
//     thread = channel).  Wave-level shuffles + tiny LDS combine (2 barriers).
// ---------------------------------------------------------------------------
__global__ void k3_scan(const float* __restrict__ inp, const float* __restrict__ invn,
                        const float* __restrict__ vmax, const int* __restrict__ ind,
                        const int* __restrict__ mask_idx, const int* __restrict__ known_idx,
                        float* __restrict__ a_new, float* __restrict__ a_ori) {
    __shared__ float red[8];
    int b    = blockIdx.x;
    int c    = threadIdx.x;
    int lane = c & 31;
    int wid  = c >> 5;
    const float* base = inp + (size_t)b * CHW + (size_t)c * HW;
    float out_c = 0.f;

    for (int t = 0; t < PM; ++t) {
        int mp = mask_idx[t];
        float u = base[mp] * invn[b * HW + mp];
        float part = out_c * u;
        #pragma unroll
        for (int m = 16; m >= 1; m >>= 1) part += __shfl_xor(part, m, 32);
        if (lane == 0) red[wid] = part;
        __syncthreads();
        float at = red[0] + red[1] + red[2] + red[3]
                 + red[4] + red[5] + red[6] + red[7];
        __syncthreads();                 // protect red[] before next write

        float vm    = vmax[b * HW + mp];
        float denom = at + vm;
        float an    = at / denom;
        float ao    = vm / denom;
        int   ch    = ind[b * HW + mp];
        float bk    = base[known_idx[ch]];
        out_c = an * out_c + ao * bk;
        if (c == 0) { a_new[b * PM + t] = an; a_ori[b * PM + t] = ao; }
    }
}

// ---------------------------------------------------------------------------
// K4: build lower-triangular M[t][s] = a_ori[s] * prod_{r=s+1..t} a_new[r]
//     (zero only the in-tile upper wedge; GEMM2 never reads past the tile)
// ---------------------------------------------------------------------------
__global__ void k4_build_M(const float* __restrict__ a_new, const float* __restrict__ a_ori,
                           float* __restrict__ M) {
    int b = blockIdx.x >> 2;
    int t = (blockIdx.x & 3) * 256 + threadIdx.x;        // 0..1023
    const float* an = a_new + b * PM;
    const float* ao = a_ori + b * PM;
    float* row = M + (size_t)b * PM * PM + (size_t)t * PM;
    int tileEnd = ((t >> 4) + 1) << 4;
    for (int s = t + 1; s < tileEnd; ++s) row[s] = 0.f;
    float w = 1.f;
    for (int s = t; s >= 0; --s) {
        row[s] = ao[s] * w;
        w *= an[s];
    }
}

// ---------------------------------------------------------------------------
// K5: pack best_known, s-fastest:  BKT[b][c][s] = inp[b, c, known_idx[ind[mask_idx[s]]]]
// ---------------------------------------------------------------------------
__global__ void k5_pack_bkT(const float* __restrict__ inp, const int* __restrict__ ind,
                            const int* __restrict__ mask_idx, const int* __restrict__ known_idx,
                            float* __restrict__ BKT) {
    int idx = blockIdx.x * blockDim.x + threadIdx.x;     // 4*256*1024
    if (idx >= BZ * CH * PM) return;
    int s = idx % PM;
    int c = (idx / PM) % CH;
    int b = idx / (PM * CH);
    int kp = known_idx[ind[b * HW + mask_idx[s]]];
    BKT[idx] = inp[(size_t)b * CHW + (size_t)c * HW + kp];
}

// ---------------------------------------------------------------------------
// K6: GEMM2 (masked rows):  R(1024x256) = M(1024x1024, lower-tri) @ BK(1024x256)
//     out[b, c, mask_idx[t]] = R[t][c].  One wave per 16x16 tile; the s-loop
//     stops at the tile diagonal (M is triangular).
// ---------------------------------------------------------------------------
__global__ void __launch_bounds__(128) k6_att_gemm(
        const float* __restrict__ M, const float* __restrict__ BKT,
        const int* __restrict__ mask_idx, float* __restrict__ out) {
    int wave = blockIdx.x * (blockDim.x >> 5) + (threadIdx.x >> 5); // 4096 waves
    int lane = threadIdx.x & 31;
    int b    = wave >> 10;
    int tile = wave & 1023;        // 64 t-blocks x 16 c-blocks
    int t0   = (tile >> 4) * 16;
    int c0   = (tile & 15) * 16;

    const float* Mb  = M   + (size_t)b * PM * PM;
    const float* BKb = BKT + (size_t)b * CH * PM;   // BKT[c*PM + s]

    int lr = lane & 15;
    int ca = (lane < 16) ? 0 : 2;

    const float* arow = Mb  + (size_t)(t0 + lr) * PM;
    const float* brow = BKb + (size_t)(c0 + lr) * PM;

    int smax = t0 + 16;            // triangular bound (multiple of 16)
    v8f acc = {};
    #pragma unroll 8
    for (int s = 0; s < smax; s += 4) {
        v2f a  = *(const v2f*)&arow[s + ca];
        v2f bb = *(const v2f*)&brow[s + ca];
        acc = __builtin_amdgcn_wmma_f32_16x16x4_f32(
            false, a, false, bb, (short)0, acc, false, false);
    }

    int rbase = t0 + ((lane < 16) ? 0 : 8);
    int c     = c0 + lr;
    float* ob = out + (size_t)b * CHW;
    #pragma unroll
    for (int i = 0; i < 8; ++i) {
        ob[(size_t)c * HW + mask_idx[rbase + i]] = acc[i];
    }
}

// ---------------------------------------------------------------------------
// K7: unmasked rows are one-hot gathers: out[b,c,kp] = inp[b,c,known_idx[ind[kp]]]
// ---------------------------------------------------------------------------
__global__ void k7_gather(const float* __restrict__ inp, const int* __restrict__ ind,
                          const int* __restrict__ known_idx, float* __restrict__ out) {
    int idx = blockIdx.x * blockDim.x + threadIdx.x;     // 4*256*3072
    if (idx >= BZ * CH * KN) return;
    int j = idx % KN;
    int c = (idx / KN) % CH;
    int b = idx / (KN * CH);
    int kp  = known_idx[j];
    int src = known_idx[ind[b * HW + kp]];
    out[(size_t)b * CHW + (size_t)c * HW + kp] =
        inp[(size_t)b * CHW + (size_t)c * HW + src];
}

// ---------------------------------------------------------------------------
extern "C" void kernel_launch(void* const* d_in, const int* in_sizes, int n_in,
                              void* d_out, int out_size, void* d_ws, size_t ws_size,
                              hipStream_t stream) {
    const float* inp       = (const float*)d_in[0];
    const float* rf        = (const float*)d_in[1];
    const int*   mask_idx  = (const int*)d_in[2];
    const int*   known_idx = (const int*)d_in[3];
    float*       out       = (float*)d_out;

    char* ws = (char*)d_ws;
    float* invn  = (float*)(ws);                                  // 64 KB
    float* vmax  = (float*)(ws + 65536);                          // 64 KB
    int*   ind   = (int*)  (ws + 131072);                         // 64 KB
    float* a_new = (float*)(ws + 196608);                         // 16 KB
    float* a_ori = (float*)(ws + 212992);                         // 16 KB
    size_t off   = 229376;
    float* knT   = (float*)(ws + off); off += (size_t)BZ*KN*CH*4; // 12 MB
    float* rfT   = (float*)(ws + off); off += (size_t)BZ*HW*CH*4; // 16 MB
    float* Mmat  = (float*)(ws + off); off += (size_t)BZ*PM*PM*4; // 16 MB
    float* BKT   = (float*)(ws + off);                            //  4 MB

    // K0: column norms
    k0_norms<<<(BZ * HW + 255) / 256, 256, 0, stream>>>(inp, invn);
    // K1: pack normalized known (c-fastest)
    k1_pack_knT<<<(BZ * KN * CH + 255) / 256, 256, 0, stream>>>(inp, invn, known_idx, knT);
    // K1b: transpose reference (c-fastest)
    k1b_pack_rfT<<<(BZ * HW * CH + 255) / 256, 256, 0, stream>>>(rf, rfT);
    // K2: WMMA GEMM1 + fused argmax (1024 waves, 2 waves/block, LDS A-panels)
    k2_sim_argmax<<<512, 64, 0, stream>>>(rfT, knT, vmax, ind);
    // K3: sequential scan (one block per image)
    k3_scan<<<BZ, CH, 0, stream>>>(inp, invn, vmax, ind, mask_idx, known_idx, a_new, a_ori);
    // K4: build triangular coefficient matrix
    k4_build_M<<<BZ * 4, 256, 0, stream>>>(a_new, a_ori, Mmat);
    // K5: pack best_known (s-fastest)
    k5_pack_bkT<<<(BZ * CH * PM + 255) / 256, 256, 0, stream>>>(inp, ind, mask_idx, known_idx, BKT);
    // K6: WMMA GEMM2 -> masked output columns (4096 waves, 4 waves/block)
    k6_att_gemm<<<1024, 128, 0, stream>>>(Mmat, BKT, mask_idx, out);
    // K7: one-hot gather -> unmasked output columns
    k7_gather<<<(BZ * CH * KN + 255) / 256, 256, 0, stream>>>(inp, ind, known_idx, out);
}